// DotProductAttention_14869176779206
// MI455X (gfx1250) — compile-verified
//
#include <hip/hip_runtime.h>

typedef __attribute__((ext_vector_type(2))) float v2f;
typedef __attribute__((ext_vector_type(4))) float v4f;
typedef __attribute__((ext_vector_type(8))) float v8f;

#define BATCH       8
#define SEQ_Q       2048
#define SEQ_V       2048
#define DIM         64
#define WAVES_PER_WG 4
#define QTILES_PER_B (SEQ_Q / 16)   // 128
#define NCHUNK       (SEQ_V / 16)   // 128
#define PROW         20             // padded LDS row stride in floats (bank-conflict-free)

__device__ __forceinline__ v8f wmma_f32(v2f a, v2f b, v8f c) {
  // D = A(16x4 f32) * B(4x16 f32) + C(16x16 f32)
  return __builtin_amdgcn_wmma_f32_16x16x4_f32(
      /*neg_a=*/false, a, /*neg_b=*/false, b,
      /*c_mod=*/(short)0, c, /*reuse_a=*/false, /*reuse_b=*/false);
}

__global__ __launch_bounds__(WAVES_PER_WG * 32)
void attn_fwd_kernel(const float* __restrict__ Q, const float* __restrict__ V,
                     float* __restrict__ ctx_out, float* __restrict__ attn_out) {
  __shared__ float pbuf[WAVES_PER_WG][16 * PROW];

  const int lane = threadIdx.x & 31;
  const int wave = threadIdx.x >> 5;
  const int gw   = blockIdx.x * WAVES_PER_WG + wave;   // global wave id = q-tile id
  const int b    = gw / QTILES_PER_B;
  const int qt   = gw % QTILES_PER_B;
  const int q0   = qt * 16;

  const int n = lane & 15;   // N index (col for B/C/D, row M for A fragments)
  const int h = lane >> 4;   // half-wave select (K-pair / M+8 select)

  const float* Qb = Q + ((size_t)b * SEQ_Q + q0) * DIM;
  const float* Vb = V + (size_t)b * SEQ_V * DIM;

  // ---- preload Q A-fragments: qa[k] holds Q[row=n][4k+2h : 4k+2h+1] ----
  v2f qa[16];
#pragma unroll
  for (int k = 0; k < 16; ++k)
    qa[k] = *(const v2f*)(Qb + n * DIM + 4 * k + 2 * h);

  // ======== Pass A: streaming per-lane (max, sumexp) over all V chunks ========
  float mrow[8], lrow[8];
#pragma unroll
  for (int r = 0; r < 8; ++r) { mrow[r] = -1e30f; lrow[r] = 0.0f; }

  for (int c = 0; c < NCHUNK; ++c) {
    const float* Vc = Vb + (size_t)c * 16 * DIM;
    if (c + 1 < NCHUNK)
      __builtin_prefetch(Vb + (size_t)(c + 1) * 16 * DIM + n * DIM, 0, 1);

    // two independent WMMA accumulator chains (halve serial RAW depth)
    v8f s0 = {0.f, 0.f, 0.f, 0.f, 0.f, 0.f, 0.f, 0.f};
    v8f s1 = {0.f, 0.f, 0.f, 0.f, 0.f, 0.f, 0.f, 0.f};
#pragma unroll
    for (int k = 0; k < 16; k += 2) {
      // B fragment of V^T: B[kk][n] = V[v0+n][4k+kk] -> lane holds kk = 2h, 2h+1
      v2f bf0 = *(const v2f*)(Vc + n * DIM + 4 * k + 2 * h);
      v2f bf1 = *(const v2f*)(Vc + n * DIM + 4 * (k + 1) + 2 * h);
      s0 = wmma_f32(qa[k],     bf0, s0);
      s1 = wmma_f32(qa[k + 1], bf1, s1);
    }
    s0 += s1;
#pragma unroll
    for (int r = 0; r < 8; ++r) {
      float v  = s0[r];
      float m  = mrow[r];
      float e  = __expf(fminf(m - v, v - m));          // exp(-|m - v|), 1 transcendental
      lrow[r]  = (v > m) ? __fmaf_rn(lrow[r], e, 1.0f) // new max: rescale old sum, +exp(0)
                         : (lrow[r] + e);              // old max kept: add exp(v-m)
      mrow[r]  = fmaxf(m, v);
    }
  }
  // cross-lane softmax-combine across the 16 lanes sharing h (xor bits 0..3 only)
#pragma unroll
  for (int off = 1; off < 16; off <<= 1) {
#pragma unroll
    for (int r = 0; r < 8; ++r) {
      float mo = __shfl_xor(mrow[r], off, 32);
      float lo = __shfl_xor(lrow[r], off, 32);
      float mn = fmaxf(mrow[r], mo);
      lrow[r]  = lrow[r] * __expf(mrow[r] - mn) + lo * __expf(mo - mn);
      mrow[r]  = mn;
    }
  }
  float inv_l[8];
#pragma unroll
  for (int r = 0; r < 8; ++r) inv_l[r] = 1.0f / lrow[r];

  // ======== Pass B: recompute S, normalize, emit attn, accumulate context ====
  v8f zero8 = {0.f, 0.f, 0.f, 0.f, 0.f, 0.f, 0.f, 0.f};
  v8f ctx[4] = {zero8, zero8, zero8, zero8};   // 4 N-tiles of 16 covering D=64

  float* my_p = &pbuf[wave][0];
  float* attn_base = attn_out + ((size_t)b * SEQ_Q + q0) * SEQ_V;

  // b128 attn-store lane mapping: ids 0..63 -> (row = id>>2, 4-float block = id&3)
  const int id0 = lane,      row0 = id0 >> 2, blk0 = id0 & 3;
  const int id1 = lane + 32, row1 = id1 >> 2, blk1 = id1 & 3;

  for (int c = 0; c < NCHUNK; ++c) {
    const float* Vc = Vb + (size_t)c * 16 * DIM;
    if (c + 1 < NCHUNK)
      __builtin_prefetch(Vb + (size_t)(c + 1) * 16 * DIM + n * DIM, 0, 1);

    v8f s0 = {0.f, 0.f, 0.f, 0.f, 0.f, 0.f, 0.f, 0.f};
    v8f s1 = {0.f, 0.f, 0.f, 0.f, 0.f, 0.f, 0.f, 0.f};
#pragma unroll
    for (int k = 0; k < 16; k += 2) {
      v2f bf0 = *(const v2f*)(Vc + n * DIM + 4 * k + 2 * h);
      v2f bf1 = *(const v2f*)(Vc + n * DIM + 4 * (k + 1) + 2 * h);
      s0 = wmma_f32(qa[k],     bf0, s0);
      s1 = wmma_f32(qa[k + 1], bf1, s1);
    }
    s0 += s1;

    // normalize; stage P tile to LDS (C/D layout -> row-major)
#pragma unroll
    for (int r = 0; r < 8; ++r) {
      float p = __expf(s0[r] - mrow[r]) * inv_l[r];
      my_p[(r + 8 * h) * PROW + n] = p;          // C/D layout: M = r + 8*h
    }
    asm volatile("s_wait_dscnt 0" ::: "memory");  // P visible before re-reads

    // attn out: two fully-coalesced b128 stores per lane (read back row-major)
    v4f p0 = *(const v4f*)(my_p + row0 * PROW + blk0 * 4);
    v4f p1 = *(const v4f*)(my_p + row1 * PROW + blk1 * 4);
    *(v4f*)(attn_base + (size_t)row0 * SEQ_V + c * 16 + blk0 * 4) = p0;
    *(v4f*)(attn_base + (size_t)row1 * SEQ_V + c * 16 + blk1 * 4) = p1;

    // ctx[t] += P[:, kb:kb+4] x Vc[kb:kb+4, 16t:16t+16]
#pragma unroll
    for (int kb = 0; kb < 16; kb += 4) {
      v2f pa = *(const v2f*)(my_p + n * PROW + kb + 2 * h);  // A frag of P
#pragma unroll
      for (int t = 0; t < 4; ++t) {
        v2f bv;
        bv.x = Vc[(kb + 2 * h)     * DIM + 16 * t + n];      // B frag of V
        bv.y = Vc[(kb + 2 * h + 1) * DIM + 16 * t + n];
        ctx[t] = wmma_f32(pa, bv, ctx[t]);
      }
    }
    asm volatile("s_wait_dscnt 0" ::: "memory");  // reads done before next overwrite
  }

  // ---- write context [16 x 64] ----
  float* ctx_base = ctx_out + ((size_t)b * SEQ_Q + q0) * DIM;
#pragma unroll
  for (int t = 0; t < 4; ++t)
#pragma unroll
    for (int r = 0; r < 8; ++r)
      ctx_base[(size_t)(r + 8 * h) * DIM + 16 * t + n] = ctx[t][r];
}

extern "C" void kernel_launch(void* const* d_in, const int* in_sizes, int n_in,
                              void* d_out, int out_size, void* d_ws, size_t ws_size,
                              hipStream_t stream) {
  const float* Q = (const float*)d_in[0];
  const float* V = (const float*)d_in[1];
  float* ctx  = (float*)d_out;                                   // [8,2048,64]
  float* attn = (float*)d_out + (size_t)BATCH * SEQ_Q * DIM;     // [8,2048,2048]

  dim3 grid((BATCH * QTILES_PER_B) / WAVES_PER_WG);  // 256 workgroups
  dim3 block(WAVES_PER_WG * 32);                     // 4 waves (wave32)
  hipLaunchKernelGGL(attn_fwd_kernel, grid, block, 0, stream, Q, V, ctx, attn);
}